// DULLI_21869973471277
// MI455X (gfx1250) — compile-verified
//
#include <hip/hip_runtime.h>
#include <hip/hip_bf16.h>

typedef __attribute__((ext_vector_type(2))) float v2f;
typedef __attribute__((ext_vector_type(8))) float v8f;

#define HW 1600
#define INV_T 10.0f

__device__ __forceinline__ v8f wmma4(v2f a, v2f b, v8f c) {
  // V_WMMA_F32_16X16X4_F32 : D = A(16x4 f32) x B(4x16 f32) + C(16x16 f32)
  return __builtin_amdgcn_wmma_f32_16x16x4_f32(false, a, false, b, (short)0, c,
                                               false, false);
}

__device__ __forceinline__ v8f zero8() {
  v8f z;
#pragma unroll
  for (int i = 0; i < 8; ++i) z[i] = 0.0f;
  return z;
}

// reductions across the 16 lanes of a half-wave (wave32; bit4 = half select)
__device__ __forceinline__ float halfMax(float v) {
#pragma unroll
  for (int m = 1; m < 16; m <<= 1) v = fmaxf(v, __shfl_xor(v, m, 32));
  return v;
}
__device__ __forceinline__ float halfSum(float v) {
#pragma unroll
  for (int m = 1; m < 16; m <<= 1) v += __shfl_xor(v, m, 32);
  return v;
}

__device__ __forceinline__ float gateOf(const int* it) {
  float g = (float)(10000 - it[0]) * 1e-4f;
  return fmaxf(g, 0.0f);
}

// ---------------------------------------------------------------- codebook |e|^2
__global__ void ee_kernel(const float* __restrict__ cb, float* __restrict__ ee,
                          int K, int C) {
  int k = blockIdx.x * blockDim.x + threadIdx.x;
  if (k >= K) return;
  const float* row = cb + (size_t)k * C;
  float s = 0.f;
  for (int c = 0; c < C; ++c) s += row[c] * row[c];
  ee[k] = s;
}

// ------------------------------------------------- distances + online softmax stats
// d[n][k] = |x_n|^2 + |e_k|^2 - 2 x_n.e_k ; also writes rowwise max/denom of
// s = -10*d for the softmax second pass.
template <int CDIM, int WAVES>
__global__ void dist_kernel(const float* __restrict__ Asrc, int aBCHW,
                            const float* __restrict__ cb,
                            const float* __restrict__ ee,
                            float* __restrict__ dOut, float* __restrict__ mOut,
                            float* __restrict__ lOut, int KC) {
  constexpr int CDP = CDIM + 1;  // LDS pad: conflict-free row reads
  __shared__ float ldsA[WAVES * 16 * CDP];
  __shared__ float xxs[WAVES * 16];
  const int lane = threadIdx.x & 31;
  const int wave = threadIdx.x >> 5;
  const int m16 = lane & 15;
  const int half = lane >> 4;
  const int rowBase = (blockIdx.x * WAVES + wave) * 16;
  float* myA = ldsA + wave * 16 * CDP;

  if (aBCHW) {  // x (B,C,H,W): lanes walk hw -> coalesced
    for (int i = lane; i < 16 * CDIM; i += 32) {
      int row = i & 15, c = i >> 4;
      int n = rowBase + row;
      myA[row * CDP + c] = Asrc[((size_t)(n / HW) * CDIM + c) * HW + (n % HW)];
    }
  } else {  // row-major (N,C): lanes walk c -> coalesced
    for (int i = lane; i < 16 * CDIM; i += 32) {
      int row = i / CDIM, c = i % CDIM;
      myA[row * CDP + c] = Asrc[(size_t)(rowBase + row) * CDIM + c];
    }
  }
  __syncthreads();

  float xx = 0.f;
  for (int c = 0; c < CDIM; ++c) {
    float v = myA[m16 * CDP + c];
    xx += v * v;
  }
  if (half == 0) xxs[wave * 16 + m16] = xx;
  __syncthreads();
  float xxr[8];
#pragma unroll
  for (int r = 0; r < 8; ++r) xxr[r] = xxs[wave * 16 + r + half * 8];

  float mr[8], lr[8];
#pragma unroll
  for (int r = 0; r < 8; ++r) { mr[r] = -3.0e38f; lr[r] = 0.f; }

  for (int kt = 0; kt < KC / 16; ++kt) {
    const int kBase = kt * 16;
    const float* cbk = cb + (size_t)(kBase + m16) * CDIM;
    v8f acc0 = zero8(), acc1 = zero8();
#pragma unroll 8
    for (int ch = 0; ch < CDIM / 4; ch += 2) {
      int c0 = ch * 4 + half * 2;
      v2f a, b;
      a.x = myA[m16 * CDP + c0];
      a.y = myA[m16 * CDP + c0 + 1];
      b.x = cbk[c0];
      b.y = cbk[c0 + 1];
      acc0 = wmma4(a, b, acc0);
      int c1 = c0 + 4;
      a.x = myA[m16 * CDP + c1];
      a.y = myA[m16 * CDP + c1 + 1];
      b.x = cbk[c1];
      b.y = cbk[c1 + 1];
      acc1 = wmma4(a, b, acc1);
    }
    v8f acc = acc0 + acc1;
    float eeK = ee[kBase + m16];
#pragma unroll
    for (int r = 0; r < 8; ++r) {
      float d = xxr[r] + eeK - 2.0f * acc[r];
      dOut[(size_t)(rowBase + r + half * 8) * KC + kBase + m16] = d;
      float s = -INV_T * d;
      float tmax = halfMax(s);
      float e = __expf(s - tmax);
      float tsum = halfSum(e);
      float nm = fmaxf(mr[r], tmax);
      lr[r] = lr[r] * __expf(mr[r] - nm) + tsum * __expf(tmax - nm);
      mr[r] = nm;
    }
  }
  if (m16 == 0) {
#pragma unroll
    for (int r = 0; r < 8; ++r) {
      mOut[rowBase + r + half * 8] = mr[r];
      lOut[rowBase + r + half * 8] = lr[r];
    }
  }
}

// --------------------------------- stage-1: prob, a0, q=prob@vq0, gate blend
__global__ void prob1_kernel(const float* __restrict__ dIn,
                             const float* __restrict__ mIn,
                             const float* __restrict__ lIn,
                             const float* __restrict__ cb,
                             const float* __restrict__ x0,
                             const int* __restrict__ curIter,
                             float* __restrict__ aOut, float* __restrict__ qxB,
                             float* __restrict__ qxRM) {
  const int lane = threadIdx.x & 31, wave = threadIdx.x >> 5;
  const int m16 = lane & 15, half = lane >> 4;
  const int rowBase = (blockIdx.x * 4 + wave) * 16;
  const int nMe = rowBase + m16;  // A-layout row owned by this lane
  const float mrow = mIn[nMe];
  const float linv = 1.0f / lIn[nMe];
  const float* drow = dIn + (size_t)nMe * 2048;
  const int bI = nMe / HW, hw = nMe % HW;
  v8f acc[4];
#pragma unroll
  for (int ct = 0; ct < 4; ++ct) acc[ct] = zero8();

  for (int kt = 0; kt < 128; ++kt) {
    const int kBase = kt * 16;
    v2f a[4];
#pragma unroll
    for (int ch = 0; ch < 4; ++ch) {
      int k0 = kBase + ch * 4 + half * 2;
      float p0 = __expf(-INV_T * drow[k0] - mrow) * linv;
      float p1 = __expf(-INV_T * drow[k0 + 1] - mrow) * linv;
      a[ch].x = p0;
      a[ch].y = p1;
      aOut[((size_t)bI * 2048 + k0) * HW + hw] = p0;
      aOut[((size_t)bI * 2048 + k0 + 1) * HW + hw] = p1;
    }
#pragma unroll
    for (int ct = 0; ct < 4; ++ct) {
#pragma unroll
      for (int ch = 0; ch < 4; ++ch) {
        int kk = kBase + ch * 4 + half * 2;
        v2f b;
        b.x = cb[(size_t)kk * 64 + ct * 16 + m16];
        b.y = cb[(size_t)(kk + 1) * 64 + ct * 16 + m16];
        acc[ct] = wmma4(a[ch], b, acc[ct]);
      }
    }
  }
  const float g = gateOf(curIter);
#pragma unroll
  for (int ct = 0; ct < 4; ++ct) {
#pragma unroll
    for (int r = 0; r < 8; ++r) {
      int n = rowBase + r + half * 8;
      int c = ct * 16 + m16;
      size_t bchw = ((size_t)(n / HW) * 64 + c) * HW + (n % HW);
      float o = x0[bchw] * g + acc[ct][r] * (1.0f - g);
      qxB[bchw] = o;
      qxRM[(size_t)n * 64 + c] = o;
    }
  }
}

// ------------------------------------------------ head: relu(x@W1^T+b1)@W2^T+b2
__global__ void head_kernel(const float* __restrict__ qxRM,
                            const float* __restrict__ w1,
                            const float* __restrict__ b1,
                            const float* __restrict__ w2,
                            const float* __restrict__ b2,
                            float* __restrict__ x3B, float* __restrict__ x3RM) {
  __shared__ float a0s[2][16 * 65];
  __shared__ float y1s[2][16 * 257];
  const int lane = threadIdx.x & 31, wave = threadIdx.x >> 5;
  const int m16 = lane & 15, half = lane >> 4;
  const int rowBase = (blockIdx.x * 2 + wave) * 16;

  for (int i = lane; i < 16 * 64; i += 32) {
    int row = i >> 6, c = i & 63;
    a0s[wave][row * 65 + c] = qxRM[(size_t)(rowBase + row) * 64 + c];
  }
  __syncthreads();

#pragma unroll 1
  for (int ct = 0; ct < 16; ++ct) {
    v8f acc = zero8();
    const float* w1r = w1 + (size_t)(ct * 16 + m16) * 64;
#pragma unroll
    for (int ch = 0; ch < 16; ++ch) {
      int c0 = ch * 4 + half * 2;
      v2f a, b;
      a.x = a0s[wave][m16 * 65 + c0];
      a.y = a0s[wave][m16 * 65 + c0 + 1];
      b.x = w1r[c0];
      b.y = w1r[c0 + 1];
      acc = wmma4(a, b, acc);
    }
    float bias = b1[ct * 16 + m16];
#pragma unroll
    for (int r = 0; r < 8; ++r) {
      float y = fmaxf(acc[r] + bias, 0.0f);
      y1s[wave][(r + half * 8) * 257 + ct * 16 + m16] = y;  // D-layout -> LDS
    }
  }
  __syncthreads();

#pragma unroll 1
  for (int ct = 0; ct < 16; ++ct) {
    v8f acc0 = zero8(), acc1 = zero8();
    const float* w2r = w2 + (size_t)(ct * 16 + m16) * 256;
#pragma unroll 8
    for (int ch = 0; ch < 64; ch += 2) {
      int c0 = ch * 4 + half * 2;
      v2f a, b;
      a.x = y1s[wave][m16 * 257 + c0];
      a.y = y1s[wave][m16 * 257 + c0 + 1];
      b.x = w2r[c0];
      b.y = w2r[c0 + 1];
      acc0 = wmma4(a, b, acc0);
      int c1 = c0 + 4;
      a.x = y1s[wave][m16 * 257 + c1];
      a.y = y1s[wave][m16 * 257 + c1 + 1];
      b.x = w2r[c1];
      b.y = w2r[c1 + 1];
      acc1 = wmma4(a, b, acc1);
    }
    v8f acc = acc0 + acc1;
    float bias = b2[ct * 16 + m16];
#pragma unroll
    for (int r = 0; r < 8; ++r) {
      int n = rowBase + r + half * 8;
      int c = ct * 16 + m16;
      float v = acc[r] + bias;
      x3B[((size_t)(n / HW) * 256 + c) * HW + (n % HW)] = v;
      x3RM[(size_t)n * 256 + c] = v;
    }
  }
}

// --------------------------------- stage-2: prob, a3, q=prob@vq1, gate blend
__global__ void prob2_kernel(const float* __restrict__ dIn,
                             const float* __restrict__ mIn,
                             const float* __restrict__ lIn,
                             const float* __restrict__ cb,
                             const float* __restrict__ x3RM,
                             const int* __restrict__ curIter,
                             float* __restrict__ aOut,
                             float* __restrict__ qxB) {
  __shared__ float pS[16 * 513];  // stride 513 -> conflict-free
  const int lane = threadIdx.x & 31;
  const int m16 = lane & 15, half = lane >> 4;
  const int rowBase = blockIdx.x * 16;
  const int nMe = rowBase + m16;
  const float mrow = mIn[nMe];
  const float linv = 1.0f / lIn[nMe];
  const float* drow = dIn + (size_t)nMe * 512;
  const int bI = nMe / HW, hw = nMe % HW;

  for (int kt = 0; kt < 32; ++kt) {
#pragma unroll
    for (int ch = 0; ch < 4; ++ch) {
      int k0 = kt * 16 + ch * 4 + half * 2;
      float p0 = __expf(-INV_T * drow[k0] - mrow) * linv;
      float p1 = __expf(-INV_T * drow[k0 + 1] - mrow) * linv;
      pS[m16 * 513 + k0] = p0;
      pS[m16 * 513 + k0 + 1] = p1;
      aOut[((size_t)bI * 512 + k0) * HW + hw] = p0;
      aOut[((size_t)bI * 512 + k0 + 1) * HW + hw] = p1;
    }
  }
  __syncthreads();

  const float g = gateOf(curIter);
#pragma unroll 1
  for (int ct = 0; ct < 16; ++ct) {
    v8f acc0 = zero8(), acc1 = zero8();
#pragma unroll 4
    for (int kt = 0; kt < 32; ++kt) {
#pragma unroll
      for (int ch = 0; ch < 4; ch += 2) {
        int k0 = kt * 16 + ch * 4 + half * 2;
        v2f a, b;
        a.x = pS[m16 * 513 + k0];
        a.y = pS[m16 * 513 + k0 + 1];
        b.x = cb[(size_t)k0 * 256 + ct * 16 + m16];
        b.y = cb[(size_t)(k0 + 1) * 256 + ct * 16 + m16];
        acc0 = wmma4(a, b, acc0);
        int k1 = k0 + 4;
        a.x = pS[m16 * 513 + k1];
        a.y = pS[m16 * 513 + k1 + 1];
        b.x = cb[(size_t)k1 * 256 + ct * 16 + m16];
        b.y = cb[(size_t)(k1 + 1) * 256 + ct * 16 + m16];
        acc1 = wmma4(a, b, acc1);
      }
    }
    v8f acc = acc0 + acc1;
#pragma unroll
    for (int r = 0; r < 8; ++r) {
      int n = rowBase + r + half * 8;
      int c = ct * 16 + m16;
      float o = x3RM[(size_t)n * 256 + c] * g + acc[r] * (1.0f - g);
      qxB[((size_t)(n / HW) * 256 + c) * HW + (n % HW)] = o;
    }
  }
}

extern "C" void kernel_launch(void* const* d_in, const int* in_sizes, int n_in,
                              void* d_out, int out_size, void* d_ws,
                              size_t ws_size, hipStream_t stream) {
  (void)in_sizes; (void)n_in; (void)out_size; (void)ws_size;
  const float* x0 = (const float*)d_in[0];
  const float* vq0 = (const float*)d_in[1];
  const float* vq1 = (const float*)d_in[2];
  const float* w1 = (const float*)d_in[3];
  const float* b1 = (const float*)d_in[4];
  const float* w2 = (const float*)d_in[5];
  const float* b2 = (const float*)d_in[6];
  const int* curIter = (const int*)d_in[7];

  const size_t N = 51200;  // 32*40*40
  float* out = (float*)d_out;  // tuple: x3, qx0, qx3, a0, a3, d0, d3
  float* x3_o = out;
  float* qx0_o = x3_o + N * 256;
  float* qx3_o = qx0_o + N * 64;
  float* a0_o = qx3_o + N * 256;
  float* a3_o = a0_o + N * 2048;
  float* d0_o = a3_o + N * 512;
  float* d3_o = d0_o + N * 2048;

  float* ws = (float*)d_ws;
  float* m0 = ws;
  float* l0 = ws + N;
  float* m3 = ws + 2 * N;
  float* l3 = ws + 3 * N;
  float* ee0 = ws + 4 * N;
  float* ee1 = ee0 + 2048;
  float* qx0rm = ee1 + 512;       // N*64  row-major qx0
  float* x3rm = qx0rm + N * 64;   // N*256 row-major x3

  ee_kernel<<<8, 256, 0, stream>>>(vq0, ee0, 2048, 64);
  ee_kernel<<<2, 256, 0, stream>>>(vq1, ee1, 512, 256);
  dist_kernel<64, 4><<<800, 128, 0, stream>>>(x0, 1, vq0, ee0, d0_o, m0, l0, 2048);
  prob1_kernel<<<800, 128, 0, stream>>>(d0_o, m0, l0, vq0, x0, curIter, a0_o,
                                        qx0_o, qx0rm);
  head_kernel<<<1600, 64, 0, stream>>>(qx0rm, w1, b1, w2, b2, x3_o, x3rm);
  dist_kernel<256, 2><<<1600, 64, 0, stream>>>(x3rm, 0, vq1, ee1, d3_o, m3, l3, 512);
  prob2_kernel<<<3200, 32, 0, stream>>>(d3_o, m3, l3, vq1, x3rm, curIter, a3_o,
                                        qx3_o);
}